// DeepEquilibriumModel_46943992546044
// MI455X (gfx1250) — compile-verified
//
#include <hip/hip_runtime.h>
#include <math.h>
#include <stdint.h>

// ---------------------------------------------------------------------------
// DEQ with Anderson acceleration on MI455X (gfx1250, wave32).
// Dominant cost: fp32 GEMMs -> V_WMMA_F32_16X16X4_F32 matrix pipe.
// Working set (~90MB) fits in the 192MB L2; compute-bound in fp32.
// This revision: transposed-B LDS layout (mov-free v2f fragments for both
// operands), async global->LDS staging of A (ASYNCcnt), double-buffered
// LDS tiles, one barrier per K-chunk.
// ---------------------------------------------------------------------------

typedef float v2f __attribute__((ext_vector_type(2)));
typedef float v8f __attribute__((ext_vector_type(8)));

constexpr int    Bsz  = 2;
constexpr size_t FLAT = 1024 * 512;         // S*D per batch
constexpr size_t NTOT = (size_t)Bsz * FLAT; // 1,048,576 elements of z
constexpr int    MM   = 2048;               // B*S rows
constexpr int    DD   = 512;
constexpr int    FF   = 2048;
constexpr float  LAM  = 1e-4f;

#define TM 128
#define TN 128
#define TK 16
#define LPAD 4    // row stride 20 floats: 8B-aligned v2f frags, conflict-free banks

__device__ __forceinline__ void async_load_b128(unsigned lds_off, const void* gaddr) {
  asm volatile("global_load_async_to_lds_b128 %0, %1, off"
               :: "v"(lds_off), "v"((unsigned long long)(uintptr_t)gaddr)
               : "memory");
}

__device__ __forceinline__ void async_wait0() {
#if __has_builtin(__builtin_amdgcn_s_wait_asynccnt)
  __builtin_amdgcn_s_wait_asynccnt(0);
#else
  asm volatile("s_wait_asynccnt 0" ::: "memory");
#endif
}

// ------------------------------ GEMM ---------------------------------------
// C[M,N] = op( A[M,K] * W[K,N] + addend ),  addend = bias[N] or matrix[M,N]
// Workgroup: 256 threads (8 waves), tile 128x128, K-chunk 16, double-buffered.
// Wave tile: 32(M) x 64(N) = 2x4 fragments of v_wmma_f32_16x16x4_f32.
//
// WMMA f32 16x16x4 per-lane layout (wave32):
//   A (16x4): lane l holds A[l%16][2*(l/16) + {0,1}]          -> v2f
//   B (4x16): lane l holds B[2*(l/16) + {0,1}][l%16]          -> v2f
//   C (16x16): VGPR r, lane l -> C[r + 8*(l/16)][l%16]        -> v8f
// Both LDS tiles are stored with K innermost (B transposed), so each
// fragment is one contiguous 8-byte ds load -> no repacking movs.

__global__ __launch_bounds__(256) void gemm_f32_wmma(
    const float* __restrict__ A, const float* __restrict__ W,
    const float* __restrict__ add, float* __restrict__ C,
    int M, int N, int K, int add_is_matrix, int do_tanh)
{
  __shared__ __align__(16) float As [2][TM][TK + LPAD];  // A tile, K innermost
  __shared__ __align__(16) float BsT[2][TN][TK + LPAD];  // B tile transposed

  const int tid  = threadIdx.x;
  const int lane = tid & 31;
  const int wv   = tid >> 5;          // 0..7
  const int wm   = wv >> 1;           // 0..3  (M direction)
  const int wn   = wv & 1;            // 0..1  (N direction)
  const int half = lane >> 4;         // 0/1
  const int l16  = lane & 15;
  const int bm   = blockIdx.y * TM;
  const int bn   = blockIdx.x * TN;

  v8f acc[2][4];
#pragma unroll
  for (int i = 0; i < 2; ++i)
#pragma unroll
    for (int j = 0; j < 4; ++j) acc[i][j] = v8f{};

  // --- staging helpers (inlined below) ---
  // A tile: 128x16 floats = 512 float4; 2 async b128 per thread.
  // B tile: 16x128 floats, loaded as float4 rows, scattered transposed.

  const int nch = K / TK;

  // Prologue: stage chunk 0.
  {
#pragma unroll
    for (int t = 0; t < 2; ++t) {
      int idx = t * 256 + tid;
      int r = idx >> 2, c = (idx & 3) << 2;
      async_load_b128((unsigned)(uintptr_t)&As[0][r][c],
                      A + (size_t)(bm + r) * K + c);
    }
    float4 br[2];
#pragma unroll
    for (int t = 0; t < 2; ++t) {
      int idx = t * 256 + tid;
      int r = idx >> 5, c = (idx & 31) << 2;
      br[t] = *reinterpret_cast<const float4*>(W + (size_t)r * N + bn + c);
    }
#pragma unroll
    for (int t = 0; t < 2; ++t) {
      int idx = t * 256 + tid;
      int r = idx >> 5, c = (idx & 31) << 2;
      BsT[0][c + 0][r] = br[t].x;
      BsT[0][c + 1][r] = br[t].y;
      BsT[0][c + 2][r] = br[t].z;
      BsT[0][c + 3][r] = br[t].w;
    }
    async_wait0();
    __syncthreads();
  }

  for (int ch = 0; ch < nch; ++ch) {
    const int cur = ch & 1;
    const int nxt = cur ^ 1;
    const int k1  = (ch + 1) * TK;
    float4 br[2];

    // Prefetch chunk ch+1: async A -> LDS buffer nxt, B -> registers.
    if (ch + 1 < nch) {
#pragma unroll
      for (int t = 0; t < 2; ++t) {
        int idx = t * 256 + tid;
        int r = idx >> 2, c = (idx & 3) << 2;
        async_load_b128((unsigned)(uintptr_t)&As[nxt][r][c],
                        A + (size_t)(bm + r) * K + k1 + c);
      }
#pragma unroll
      for (int t = 0; t < 2; ++t) {
        int idx = t * 256 + tid;
        int r = idx >> 5, c = (idx & 31) << 2;
        br[t] = *reinterpret_cast<const float4*>(W + (size_t)(k1 + r) * N + bn + c);
      }
    }

    // Compute chunk ch: 4 K-steps x 8 WMMA.
#pragma unroll
    for (int kk = 0; kk < 4; ++kk) {
      const int kb = kk * 4 + half * 2;
      v2f a[2], b[4];
#pragma unroll
      for (int i = 0; i < 2; ++i)
        a[i] = *reinterpret_cast<const v2f*>(&As[cur][wm * 32 + i * 16 + l16][kb]);
#pragma unroll
      for (int j = 0; j < 4; ++j)
        b[j] = *reinterpret_cast<const v2f*>(&BsT[cur][wn * 64 + j * 16 + l16][kb]);
#pragma unroll
      for (int i = 0; i < 2; ++i)
#pragma unroll
        for (int j = 0; j < 4; ++j)
          acc[i][j] = __builtin_amdgcn_wmma_f32_16x16x4_f32(
              false, a[i], false, b[j], (short)0, acc[i][j], false, false);
    }

    // Commit B registers transposed into LDS buffer nxt; wait for async A.
    if (ch + 1 < nch) {
#pragma unroll
      for (int t = 0; t < 2; ++t) {
        int idx = t * 256 + tid;
        int r = idx >> 5, c = (idx & 31) << 2;
        BsT[nxt][c + 0][r] = br[t].x;
        BsT[nxt][c + 1][r] = br[t].y;
        BsT[nxt][c + 2][r] = br[t].z;
        BsT[nxt][c + 3][r] = br[t].w;
      }
      async_wait0();
    }
    __syncthreads();
  }

  // Epilogue: fused addend + optional tanh.
#pragma unroll
  for (int i = 0; i < 2; ++i) {
#pragma unroll
    for (int j = 0; j < 4; ++j) {
      const int col = bn + wn * 64 + j * 16 + l16;
#pragma unroll
      for (int r = 0; r < 8; ++r) {
        const int row = bm + wm * 32 + i * 16 + half * 8 + r;
        float v = acc[i][j][r];
        v += add_is_matrix ? add[(size_t)row * N + col] : add[col];
        if (do_tanh) v = tanhf(v);
        C[(size_t)row * N + col] = v;
      }
    }
  }
}

// --------------------------- Anderson machinery -----------------------------

__global__ void zero_kernel(float* p, size_t n) {
  size_t i = (size_t)blockIdx.x * blockDim.x + threadIdx.x;
  if (i < n) p[i] = 0.f;
}

// Warmup (iter < M): store history, z <- f(z)   (beta = 1)
__global__ void simple_update(float* z, const float* __restrict__ fz,
                              float* Zh, float* Gh, int cur) {
  size_t i = (size_t)blockIdx.x * blockDim.x + threadIdx.x;
  if (i >= NTOT) return;
  float zv = z[i], fv = fz[i];
  Zh[(size_t)cur * NTOT + i] = zv;
  Gh[(size_t)cur * NTOT + i] = fv - zv;
  z[i] = fv;
}

// 14 per-batch dot products: HTH upper triangle (10) + HTy (4).
__global__ __launch_bounds__(256) void anderson_dots(
    const float* __restrict__ fz, const float* __restrict__ z,
    const float* __restrict__ Gh, int s0, int s1, int s2, int s3,
    float* red)
{
  const int b = blockIdx.y;
  const size_t base = (size_t)b * FLAT;
  float s[14];
#pragma unroll
  for (int q = 0; q < 14; ++q) s[q] = 0.f;

  for (size_t f = (size_t)blockIdx.x * blockDim.x + threadIdx.x; f < FLAT;
       f += (size_t)gridDim.x * blockDim.x) {
    const size_t i = base + f;
    float g  = fz[i] - z[i];
    float d0 = g - Gh[(size_t)s0 * NTOT + i];
    float d1 = g - Gh[(size_t)s1 * NTOT + i];
    float d2 = g - Gh[(size_t)s2 * NTOT + i];
    float d3 = g - Gh[(size_t)s3 * NTOT + i];
    s[0] += d0 * d0; s[1] += d0 * d1; s[2] += d0 * d2; s[3] += d0 * d3;
    s[4] += d1 * d1; s[5] += d1 * d2; s[6] += d1 * d3;
    s[7] += d2 * d2; s[8] += d2 * d3; s[9] += d3 * d3;
    s[10] += d0 * g; s[11] += d1 * g; s[12] += d2 * g; s[13] += d3 * g;
  }

  __shared__ float sh[256];
  for (int q = 0; q < 14; ++q) {
    sh[threadIdx.x] = s[q];
    __syncthreads();
    for (int off = 128; off > 0; off >>= 1) {
      if ((int)threadIdx.x < off) sh[threadIdx.x] += sh[threadIdx.x + off];
      __syncthreads();
    }
    if (threadIdx.x == 0) atomicAdd(&red[b * 14 + q], sh[0]);
    __syncthreads();
  }
}

// gamma = (HTH + LAM*I)^-1 HTy, one lane per batch, Gauss-Jordan 4x4.
__global__ void anderson_gamma(const float* __restrict__ red, float* gamma) {
  int b = threadIdx.x;
  if (b >= Bsz) return;
  const float* r = red + b * 14;
  float Au[4][5] = {
    { r[0] + LAM, r[1],       r[2],       r[3],       r[10] },
    { r[1],       r[4] + LAM, r[5],       r[6],       r[11] },
    { r[2],       r[5],       r[7] + LAM, r[8],       r[12] },
    { r[3],       r[6],       r[8],       r[9] + LAM, r[13] },
  };
#pragma unroll
  for (int c = 0; c < 4; ++c) {
    float inv = 1.0f / (Au[c][c] + 1e-6f);
#pragma unroll
    for (int j = 0; j < 5; ++j) Au[c][j] *= inv;
#pragma unroll
    for (int rr = 0; rr < 4; ++rr) {
      if (rr == c) continue;
      float m = Au[rr][c];
#pragma unroll
      for (int j = 0; j < 5; ++j) Au[rr][j] -= m * Au[c][j];
    }
  }
#pragma unroll
  for (int i = 0; i < 4; ++i) gamma[b * 4 + i] = Au[i][4];
}

// z <- f(z) - sum_j gamma_j * (DZ_j + DG_j); store history.
__global__ void anderson_update(float* z, const float* __restrict__ fz,
                                float* Zh, float* Gh,
                                const float* __restrict__ gamma,
                                int cur, int s0, int s1, int s2, int s3) {
  size_t i = (size_t)blockIdx.x * blockDim.x + threadIdx.x;
  if (i >= NTOT) return;
  int b = (int)(i / FLAT);
  const float g0 = gamma[b * 4 + 0], g1 = gamma[b * 4 + 1];
  const float g2 = gamma[b * 4 + 2], g3 = gamma[b * 4 + 3];
  float zv = z[i], fv = fz[i];
  float g = fv - zv;
  float corr = 0.f;
  {
    float dz = zv - Zh[(size_t)s0 * NTOT + i];
    float dg = g  - Gh[(size_t)s0 * NTOT + i];
    corr += g0 * (dz + dg);
  }
  {
    float dz = zv - Zh[(size_t)s1 * NTOT + i];
    float dg = g  - Gh[(size_t)s1 * NTOT + i];
    corr += g1 * (dz + dg);
  }
  {
    float dz = zv - Zh[(size_t)s2 * NTOT + i];
    float dg = g  - Gh[(size_t)s2 * NTOT + i];
    corr += g2 * (dz + dg);
  }
  {
    float dz = zv - Zh[(size_t)s3 * NTOT + i];
    float dg = g  - Gh[(size_t)s3 * NTOT + i];
    corr += g3 * (dz + dg);
  }
  Zh[(size_t)cur * NTOT + i] = zv;
  Gh[(size_t)cur * NTOT + i] = g;
  z[i] = fv - corr;
}

// ------------------------------- Host --------------------------------------

extern "C" void kernel_launch(void* const* d_in, const int* in_sizes, int n_in,
                              void* d_out, int out_size, void* d_ws, size_t ws_size,
                              hipStream_t stream) {
  const float* x  = (const float*)d_in[0];   // (2,1024,512)
  const float* W1 = (const float*)d_in[1];   // (512,2048)
  const float* Wx = (const float*)d_in[2];   // (512,2048)
  const float* b1 = (const float*)d_in[3];   // (2048,)
  const float* W2 = (const float*)d_in[4];   // (2048,512)
  const float* b2 = (const float*)d_in[5];   // (512,)

  float* z  = (float*)d_out;                 // live state, (2048,512)
  float* ws = (float*)d_ws;

  float* xwx   = ws;                          // 2048*2048  (x@Wx + b1)
  float* h     = xwx + (size_t)MM * FF;       // 2048*2048
  float* fz    = h   + (size_t)MM * FF;       // 2048*512
  float* Zh    = fz  + NTOT;                  // 5 * NTOT
  float* Gh    = Zh  + 5 * NTOT;              // 5 * NTOT
  float* red   = Gh  + 5 * NTOT;              // 28 (pad 32)
  float* gamma = red + 32;                    // 8

  const int eb = 256;
  const int egrid = (int)((NTOT + eb - 1) / eb);

  // z = 0
  zero_kernel<<<egrid, eb, 0, stream>>>(z, NTOT);

  // xwx = x @ Wx + b1   (precomputed once; x is constant across iterations)
  gemm_f32_wmma<<<dim3(FF / TN, MM / TM), 256, 0, stream>>>(
      x, Wx, b1, xwx, MM, FF, DD, /*matrix=*/0, /*tanh=*/0);

  for (int it = 0; it < 12; ++it) {
    // h = tanh(z @ W1 + xwx)
    gemm_f32_wmma<<<dim3(FF / TN, MM / TM), 256, 0, stream>>>(
        z, W1, xwx, h, MM, FF, DD, /*matrix=*/1, /*tanh=*/1);
    // fz = h @ W2 + b2
    gemm_f32_wmma<<<dim3(DD / TN, MM / TM), 256, 0, stream>>>(
        h, W2, b2, fz, MM, DD, FF, /*matrix=*/0, /*tanh=*/0);

    const int cur = it % 5;
    if (it < 5) {
      simple_update<<<egrid, eb, 0, stream>>>(z, fz, Zh, Gh, cur);
    } else {
      const int s0 = (it - 1) % 5, s1 = (it - 2) % 5;
      const int s2 = (it - 3) % 5, s3 = (it - 4) % 5;
      zero_kernel<<<1, 64, 0, stream>>>(red, 40);
      anderson_dots<<<dim3(128, Bsz), 256, 0, stream>>>(fz, z, Gh, s0, s1, s2, s3, red);
      anderson_gamma<<<1, 32, 0, stream>>>(red, gamma);
      anderson_update<<<egrid, eb, 0, stream>>>(z, fz, Zh, Gh, gamma, cur, s0, s1, s2, s3);
    }
  }
}